// CNOT_50259707298070
// MI455X (gfx1250) — compile-verified
//
#include <hip/hip_runtime.h>

// CNOT(dim=2, wires=12, control=0, target=1) applied to x: (4096, 8192) fp32.
// U is a permutation matrix: out[row, :] = x[row ^ (((row>>11)&1) << 10), :].
// This is pure data movement: 268 MB of HBM traffic -> ~11.5 us at 23.3 TB/s.
// Any WMMA-based dense matmul would be 100x+ slower (275 GFLOP of fake work),
// so the optimized kernel is a bandwidth-saturating permuted stream copy with
// 128-bit accesses and non-temporal cache hints (working set > 192 MB L2,
// zero reuse).

typedef __attribute__((ext_vector_type(4))) float f32x4;

#define N_ROWS      4096      // 2^12 basis states
#define BATCH       8192      // columns
#define F4_PER_ROW  (BATCH / 4)      // 2048 float4 per row
#define F4_PER_BLK  1024             // half a row per block
#define THREADS     256              // 8 wave32s
#define F4_PER_THR  (F4_PER_BLK / THREADS)  // 4

__global__ __launch_bounds__(THREADS) void cnot_permute_rows(
    const float* __restrict__ x, float* __restrict__ out) {
  // Each block handles a contiguous half-row: all lanes share one src row,
  // so the CNOT permutation is scalar (SALU) and folds into the base address.
  const unsigned row      = blockIdx.x >> 1;                       // 0..4095
  const unsigned src_row  = row ^ (((row >> 11) & 1u) << 10);      // CNOT perm
  const unsigned col_base = (blockIdx.x & 1u) * F4_PER_BLK;        // 0 or 1024

  const f32x4* __restrict__ src =
      (const f32x4*)x + (size_t)src_row * F4_PER_ROW + col_base;
  f32x4* __restrict__ dst =
      (f32x4*)out + (size_t)row * F4_PER_ROW + col_base;

  const unsigned t = threadIdx.x;

  // 4x unrolled 16B loads first (4 outstanding global_load_b128 per lane),
  // then 4 stores. Non-temporal on both sides: streamed once, never reused.
  f32x4 v[F4_PER_THR];
#pragma unroll
  for (int i = 0; i < F4_PER_THR; ++i) {
    v[i] = __builtin_nontemporal_load(src + t + i * THREADS);
  }
#pragma unroll
  for (int i = 0; i < F4_PER_THR; ++i) {
    __builtin_nontemporal_store(v[i], dst + t + i * THREADS);
  }
}

extern "C" void kernel_launch(void* const* d_in, const int* in_sizes, int n_in,
                              void* d_out, int out_size, void* d_ws,
                              size_t ws_size, hipStream_t stream) {
  // setup_inputs() order: d_in[0] = U (4096*4096 f32, unused — U is a
  // permutation matrix and its action is computed analytically),
  // d_in[1] = x (4096*8192 f32).
  const float* x = (const float*)d_in[1];
  float* out = (float*)d_out;

  const int total_f4 = N_ROWS * F4_PER_ROW;          // 8,388,608
  const int blocks = total_f4 / F4_PER_BLK;          // 8192
  cnot_permute_rows<<<dim3(blocks), dim3(THREADS), 0, stream>>>(x, out);
}